// MimoAf_51436528337747
// MI455X (gfx1250) — compile-verified
//
#include <hip/hip_runtime.h>
#include <hip/hip_bf16.h>

#define TAPS 32
#define SPS 2
#define EPSC 1e-8f
#define GMAX 30.0f
#define RING_SAMPLES 2048            // LDS ring (power of 2), 4 chunks
#define CHUNK_SAMPLES 512            // samples per prefetch chunk
#define STEPS_PER_CHUNK (CHUNK_SAMPLES / SPS)   // 256 frames per chunk
#define ASYNC_PER_CHUNK 16           // 8 B128 for re + 8 for im

typedef __attribute__((ext_vector_type(4))) float v4f;  // native vector for NT store

// ---------------- complex helpers (float32, full reference precision) -------
struct cpx { float x, y; };
__device__ __forceinline__ cpx cadd(cpx a, cpx b) { return {a.x + b.x, a.y + b.y}; }
__device__ __forceinline__ cpx csub(cpx a, cpx b) { return {a.x - b.x, a.y - b.y}; }
__device__ __forceinline__ cpx cmul(cpx a, cpx b) { return {a.x * b.x - a.y * b.y, a.x * b.y + a.y * b.x}; }
__device__ __forceinline__ cpx cmulc(cpx a, cpx b) { // a * conj(b)
  return {a.x * b.x + a.y * b.y, a.y * b.x - a.x * b.y};
}
__device__ __forceinline__ cpx cconj(cpx a) { return {a.x, -a.y}; }
__device__ __forceinline__ cpx cscale(cpx a, float s) { return {a.x * s, a.y * s}; }
__device__ __forceinline__ float cnorm2(cpx a) { return a.x * a.x + a.y * a.y; }

// nearest 16QAM level per real axis: {-3,-1,1,3}/sqrt(10) (separable argmin)
__device__ __forceinline__ float qlevel(float v) {
  const float s10 = 3.1622776601683795f;    // sqrt(10)
  const float is10 = 0.31622776601683794f;  // 1/sqrt(10)
  float t = v * s10;
  float l = 2.0f * rintf((t - 1.0f) * 0.5f) + 1.0f; // nearest odd integer
  l = fminf(3.0f, fmaxf(-3.0f, l));
  return l * is10;
}
__device__ __forceinline__ cpx decide(cpx q) { return {qlevel(q.x), qlevel(q.y)}; }

// magnitude clip: g * (GMAX / max(|g|, GMAX))
__device__ __forceinline__ cpx clipg(cpx g) {
  float m = sqrtf(cnorm2(g));
  float sc = GMAX / fmaxf(m, GMAX);
  return cscale(g, sc);
}

// ---------------- CDNA5 async global -> LDS (ASYNCcnt-tracked) --------------
__device__ __forceinline__ unsigned lds_byte_addr(const void* p) {
  // flat shared address: [63:32]=aperture, [31:0]=LDS byte address (ISA 10.2)
  return (unsigned)(unsigned long long)p;
}
__device__ __forceinline__ void async_b128(unsigned lds_addr, const void* gaddr) {
  asm volatile("global_load_async_to_lds_b128 %0, %1, off"
               :: "v"(lds_addr), "v"(gaddr) : "memory");
}
__device__ __forceinline__ void wait_async_0() {
  asm volatile("s_wait_asynccnt 0x0" ::: "memory");
}
__device__ __forceinline__ void wait_async_16() {
  asm volatile("s_wait_asynccnt 0x10" ::: "memory");  // allow newest chunk in flight
}

// Copy one 512-sample chunk of (re,im) signal into the LDS ring.
// Each lane moves 2 consecutive samples (16B = B128) per issue; 8 issues x 2 arrays.
__device__ __forceinline__ void issue_chunk(const float* __restrict__ sig_re,
                                            const float* __restrict__ sig_im,
                                            float* lre, float* lim,
                                            int chunk, int lane, int nsamp) {
  int sbase = chunk * CHUNK_SAMPLES + lane * 2;
#pragma unroll
  for (int it = 0; it < CHUNK_SAMPLES / 64; ++it) {   // 8 iterations
    int s = sbase + it * 64;
    if (s + 1 < nsamp) {
      unsigned foff = (unsigned)((s & (RING_SAMPLES - 1)) * 2); // float index (dims=2)
      async_b128(lds_byte_addr(&lre[foff]), (const void*)(sig_re + (size_t)s * 2));
      async_b128(lds_byte_addr(&lim[foff]), (const void*)(sig_im + (size_t)s * 2));
    }
  }
}

// ---------------- persistent single-wave DDLMS scan --------------------------
__global__ __launch_bounds__(32) void MimoAf_ddlms_kernel(
    const float* __restrict__ sig_re, const float* __restrict__ sig_im,
    const float* __restrict__ p_eta_w, const float* __restrict__ p_eta_f,
    const float* __restrict__ p_eta_s, const float* __restrict__ p_eta_b,
    const float* __restrict__ p_beta,
    float* __restrict__ out, int nsamp) {
  __shared__ float lre[RING_SAMPLES * 2];
  __shared__ float lim[RING_SAMPLES * 2];

  const int lane = threadIdx.x;  // lane == tap index t (wave32)
  const int nframes = (nsamp - TAPS) / SPS + 1;
  const int nchunks = (nsamp + CHUNK_SAMPLES - 1) / CHUNK_SAMPLES;

  // hyperparameter transforms run on the scalar unit (off the vector path)
  const float lr_w = tanhf(p_eta_w[0]);
  const float lr_f = expf(p_eta_f[0]);
  const float lr_s = expf(p_eta_s[0]);
  const float lr_b = expf(p_eta_b[0]);
  const float beta = expf(p_beta[0]);

  // per-lane FIR weights w[i][j][t=lane]
  cpx w00{0.f, 0.f}, w01{0.f, 0.f}, w10{0.f, 0.f}, w11{0.f, 0.f};
  // replicated scalar state
  cpx f0{1.f, 0.f}, f1{1.f, 0.f}, s0{1.f, 0.f}, s1{1.f, 0.f};
  cpx b0{0.f, 0.f}, b1{0.f, 0.f}, fs0{1.f, 0.f}, fs1{1.f, 0.f};

  // prime the LDS ring: chunks 0,1 must be resident; chunk 2 in flight
  issue_chunk(sig_re, sig_im, lre, lim, 0, lane, nsamp);
  if (nchunks > 1) issue_chunk(sig_re, sig_im, lre, lim, 1, lane, nsamp);
  wait_async_0();
  if (nchunks > 2) issue_chunk(sig_re, sig_im, lre, lim, 2, lane, nsamp);

  int n = 0;
  for (int c = 0; c < nchunks && n < nframes; ++c) {
    int nEnd = (c + 1) * STEPS_PER_CHUNK;
    if (nEnd > nframes) nEnd = nframes;

    // preload u for the first frame of this chunk (chunks c, c+1 resident)
    int idx0 = ((n * SPS + lane) & (RING_SAMPLES - 1)) * 2;
    float2 re_c = *(const float2*)&lre[idx0];
    float2 im_c = *(const float2*)&lim[idx0];

    for (; n < nEnd; ++n) {
      // ---- software-pipelined prefetch of u_{n+1} (stays within this chunk's
      //      residency window: frame n+1 reads at most 30 samples into c+1) ----
      float2 re_n = re_c, im_n = im_c;
      if (n + 1 < nEnd) {
        int idx = (((n + 1) * SPS + lane) & (RING_SAMPLES - 1)) * 2;
        re_n = *(const float2*)&lre[idx];
        im_n = *(const float2*)&lim[idx];
      }
      cpx u0{re_c.x, im_c.x}, u1{re_c.y, im_c.y};

      // ---- per-lane partial products for v = einsum('ijt,tj->i', w, u) ----
      cpx p0 = cadd(cmul(w00, u0), cmul(w01, u1));
      cpx p1 = cadd(cmul(w10, u0), cmul(w11, u1));
      float pn = cnorm2(u0) + cnorm2(u1);

      // ---- wave32 butterfly reduction (all lanes end with the sums) ----
#pragma unroll
      for (int m = 16; m >= 1; m >>= 1) {
        p0.x += __shfl_xor(p0.x, m, 32);
        p0.y += __shfl_xor(p0.y, m, 32);
        p1.x += __shfl_xor(p1.x, m, 32);
        p1.y += __shfl_xor(p1.y, m, 32);
        pn   += __shfl_xor(pn,   m, 32);
      }
      cpx v0 = p0, v1 = p1;
      float usum = pn;

      // ---- scalar chain (replicated in every lane) ----
      cpx k0 = cmul(v0, f0), k1 = cmul(v1, f1);
      cpx c0 = cmul(k0, s0), c1 = cmul(k1, s1);
      cpx z0 = cadd(c0, b0), z1 = cadd(c1, b1);
      cpx q0 = cadd(cmul(v0, fs0), b0), q1 = cadd(cmul(v1, fs1), b1);
      cpx d0 = decide(q0), d1 = decide(q1);

      if (lane == 0) {  // y[n] = z  (complex64 pair, interleaved re/im); NT store
        v4f zo = {z0.x, z0.y, z1.x, z1.y};
        __builtin_nontemporal_store(zo, (v4f*)&out[4 * n]);
      }

      // psi_hat = conj(f)conj(s) / (|f||s|)
      cpx ph0 = cscale(cmul(cconj(f0), cconj(s0)), rsqrtf(cnorm2(f0) * cnorm2(s0)));
      cpx ph1 = cscale(cmul(cconj(f1), cconj(s1)), rsqrtf(cnorm2(f1) * cnorm2(s1)));

      cpx db0 = csub(d0, b0), db1 = csub(d1, b1);
      cpx ew0 = csub(cmul(db0, ph0), v0), ew1 = csub(cmul(db1, ph1), v1);
      cpx ef0 = csub(db0, k0), ef1 = csub(db1, k1);
      cpx es0 = csub(db0, c0), es1 = csub(db1, c1);
      cpx eb0 = csub(d0, z0), eb1 = csub(d1, z1);

      // ---- rank-1 weight update (per-lane taps) ----
      float gws = -1.0f / (usum + EPSC);
      cpx g;
      g = clipg(cscale(cmulc(ew0, u0), gws)); w00 = csub(w00, cscale(g, lr_w));
      g = clipg(cscale(cmulc(ew0, u1), gws)); w01 = csub(w01, cscale(g, lr_w));
      g = clipg(cscale(cmulc(ew1, u0), gws)); w10 = csub(w10, cscale(g, lr_w));
      g = clipg(cscale(cmulc(ew1, u1), gws)); w11 = csub(w11, cscale(g, lr_w));

      // ---- scalar state updates ----
      g = clipg(cscale(cmulc(ef0, v0), -1.0f / (cnorm2(v0) + EPSC))); f0 = csub(f0, cscale(g, lr_f));
      g = clipg(cscale(cmulc(ef1, v1), -1.0f / (cnorm2(v1) + EPSC))); f1 = csub(f1, cscale(g, lr_f));
      g = clipg(cscale(cmulc(es0, k0), -1.0f / (cnorm2(k0) + EPSC))); s0 = csub(s0, cscale(g, lr_s));
      g = clipg(cscale(cmulc(es1, k1), -1.0f / (cnorm2(k1) + EPSC))); s1 = csub(s1, cscale(g, lr_s));
      b0 = cadd(b0, cscale(eb0, lr_b));  // b -= lr_b * (-e_b)
      b1 = cadd(b1, cscale(eb1, lr_b));
      fs0 = cadd(cscale(fs0, beta), cscale(cmul(f0, s0), 1.0f - beta));
      fs1 = cadd(cscale(fs1, beta), cscale(cmul(f1, s1), 1.0f - beta));

      re_c = re_n; im_c = im_n;
    }

    // double-buffer schedule: slot (c+3)&3 held chunk c-1 (no longer read).
    // If we issued a new chunk, "<=16" guarantees chunk c+2 retired (in-order
    // completion). If nothing new was issued (signal tail), drain to 0 so the
    // wait cannot pass while chunk c+2 is still in flight.
    if (c + 3 < nchunks) {
      issue_chunk(sig_re, sig_im, lre, lim, c + 3, lane, nsamp);
      wait_async_16();
    } else {
      wait_async_0();
    }
  }
}

extern "C" void kernel_launch(void* const* d_in, const int* in_sizes, int n_in,
                              void* d_out, int out_size, void* d_ws, size_t ws_size,
                              hipStream_t stream) {
  const float* sig_re = (const float*)d_in[0];
  const float* sig_im = (const float*)d_in[1];
  const float* eta_w  = (const float*)d_in[2];
  const float* eta_f  = (const float*)d_in[3];
  const float* eta_s  = (const float*)d_in[4];
  const float* eta_b  = (const float*)d_in[5];
  const float* beta   = (const float*)d_in[6];
  int nsamp = in_sizes[0] / 2;  // sig_re is (NSAMP, DIMS=2)

  // serial recursion: one persistent wave32 owns the whole scan
  MimoAf_ddlms_kernel<<<1, 32, 0, stream>>>(sig_re, sig_im, eta_w, eta_f, eta_s,
                                            eta_b, beta, (float*)d_out, nsamp);
}